// IoULoss_9096740733450
// MI455X (gfx1250) — compile-verified
//
#include <hip/hip_runtime.h>
#include <hip/hip_bf16.h>

// IoU loss: -sum(log(iou + eps)) over N boxes (x, y, s) x (x, y, s).
// Memory-bound streaming reduction: 192 MB read -> ~8.2 us at 23.3 TB/s.
// Kernel 1: coalesced b128 streaming + deterministic block partial sums.
// Kernel 2: one wave folds the block partials with V_WMMA_F32_16X16X4_F32
//           (A = ones), i.e. an exact f32 tensor-core reduction.

#define IOU_EPS 1e-7f

typedef __attribute__((ext_vector_type(2))) float v2f;
typedef __attribute__((ext_vector_type(8))) float v8f;

__device__ __forceinline__ float iou_term(float ox, float oy, float os,
                                          float tx, float ty, float ts) {
    float w = fminf(ox + os, tx + ts) - fmaxf(ox - os, tx - ts);
    w = fmaxf(w, 0.0f);
    float h = fminf(oy + os, ty + ts) - fmaxf(oy - os, ty - ts);
    h = fmaxf(h, 0.0f);
    float overlap = w * h;
    float uni = 4.0f * os * os + 4.0f * ts * ts - overlap;
    float iou = overlap / (uni + IOU_EPS);
    return logf(iou + IOU_EPS);
}

__global__ __launch_bounds__(256) void iou_partial_kernel(
    const float* __restrict__ outs,   // [N,3] flattened
    const float* __restrict__ tars,   // [N,3] flattened
    float* __restrict__ partials,     // [gridDim.x]
    int nboxes) {
    const int tid = blockIdx.x * blockDim.x + threadIdx.x;
    const int tcount = gridDim.x * blockDim.x;

    float acc = 0.0f;

    // Vector main loop: 4 boxes (= 12 floats = 3 x float4) per array per iter.
    const int nvec = nboxes & ~3;
    for (int base = tid * 4; base < nvec; base += tcount * 4) {
        const float4* po = (const float4*)(outs + 3 * base);
        const float4* pt = (const float4*)(tars + 3 * base);
        float4 o0 = po[0], o1 = po[1], o2 = po[2];
        float4 t0 = pt[0], t1 = pt[1], t2 = pt[2];

        acc += iou_term(o0.x, o0.y, o0.z, t0.x, t0.y, t0.z);
        acc += iou_term(o0.w, o1.x, o1.y, t0.w, t1.x, t1.y);
        acc += iou_term(o1.z, o1.w, o2.x, t1.z, t1.w, t2.x);
        acc += iou_term(o2.y, o2.z, o2.w, t2.y, t2.z, t2.w);
    }

    // Scalar tail (N not a multiple of 4).
    for (int i = nvec + tid; i < nboxes; i += tcount) {
        acc += iou_term(outs[3 * i], outs[3 * i + 1], outs[3 * i + 2],
                        tars[3 * i], tars[3 * i + 1], tars[3 * i + 2]);
    }

    // Wave32 reduction (deterministic butterfly).
    #pragma unroll
    for (int m = 16; m >= 1; m >>= 1)
        acc += __shfl_xor(acc, m, 32);

    __shared__ float red[8];
    const int wave = threadIdx.x >> 5;
    if ((threadIdx.x & 31) == 0) red[wave] = acc;
    __syncthreads();

    if (threadIdx.x == 0) {
        float s = 0.0f;
        #pragma unroll
        for (int i = 0; i < 8; ++i) s += red[i];   // fixed order -> deterministic
        partials[blockIdx.x] = s;
    }
}

// One wave folds nparts (multiple of 64) partials with f32 WMMA:
// D = ones(16x4) x B(4x16) + C  => every D element is a column sum of B.
// Accumulating C across chunks, row 0 of D (VGPR0, lanes 0..15) holds the
// 16 column totals; a 5-step shuffle sum over those lanes gives the exact sum.
__global__ __launch_bounds__(32) void iou_final_kernel(
    const float* __restrict__ partials,
    float* __restrict__ out,
    int nparts) {
    const int lane = threadIdx.x;

    v2f a;
    a.x = 1.0f;
    a.y = 1.0f;           // A = all ones (layout-independent)
    v8f c = {};

    for (int k = 0; k < nparts; k += 64) {
        v2f b;
        b.x = partials[k + 2 * lane];
        b.y = partials[k + 2 * lane + 1];
        // 8 args: (neg_a, A, neg_b, B, c_mod, C, reuse_a, reuse_b)
        c = __builtin_amdgcn_wmma_f32_16x16x4_f32(
            false, a, false, b, (short)0, c, false, false);
    }

    // Lanes 0..15 of c[0] hold the 16 column sums (lanes 16..31 duplicate
    // them for row M=8); mask the duplicates and butterfly-reduce.
    float v = (lane < 16) ? c[0] : 0.0f;
    #pragma unroll
    for (int m = 16; m >= 1; m >>= 1)
        v += __shfl_xor(v, m, 32);

    if (lane == 0) out[0] = -v;
}

extern "C" void kernel_launch(void* const* d_in, const int* in_sizes, int n_in,
                              void* d_out, int out_size, void* d_ws, size_t ws_size,
                              hipStream_t stream) {
    const float* outs = (const float*)d_in[0];
    const float* tars = (const float*)d_in[1];
    float* partials = (float*)d_ws;
    float* out = (float*)d_out;

    const int nboxes = in_sizes[0] / 3;

    // Block count: multiple of 64 (WMMA chunk), capped by workspace.
    int nb = (int)(ws_size / sizeof(float));
    nb = (nb / 64) * 64;
    if (nb > 1024) nb = 1024;
    if (nb < 64) nb = 64;   // ws_size is expected to cover at least 256 B

    iou_partial_kernel<<<nb, 256, 0, stream>>>(outs, tars, partials, nboxes);
    iou_final_kernel<<<1, 32, 0, stream>>>(partials, out, nb);
}